// FoldNd_57363583205829
// MI455X (gfx1250) — compile-verified
//
#include <hip/hip_runtime.h>
#include <stdint.h>

// Fold (col2im): B=16, C=64, H=W=128, K=3, PAD=1, STR=DIL=1  -> LH=LW=128
#define BB     16
#define CC     64
#define HH     128
#define WW     128
#define LHD    128
#define LWD    128
#define NPLANE 9          // K*K planes per channel
#define ROWS   8          // output rows per workgroup tile
#define PITCH  136        // LDS row pitch in floats (128 data + pad; 16B-aligned rows)
#define PLANE_F (ROWS * PITCH)   // 1088 floats per plane slot

typedef __attribute__((ext_vector_type(4))) unsigned int v4u;
typedef __attribute__((ext_vector_type(8))) unsigned int v8u;
typedef __attribute__((ext_vector_type(4))) float        v4f;

__global__ __launch_bounds__(256) void fold_tdm_kernel(const float* __restrict__ x,
                                                       float* __restrict__ out) {
    // 9 planes x 8 rows x 136 floats = 39,168 bytes of LDS
    __shared__ __align__(16) float lds[NPLANE * PLANE_F];

    const unsigned bid  = blockIdx.x;
    const unsigned tile = bid & 15u;          // 16 tiles of 8 rows
    const unsigned c    = (bid >> 4) & 63u;
    const unsigned b    = bid >> 10;
    const int h0        = (int)tile * ROWS;
    const unsigned t    = threadIdx.x;

    // Plane (kh,kw) stores element lw at column lw+kw+3  ->  the tap for output w
    // is at column w+4 in EVERY plane. Only two columns ever need explicit zeros:
    //   kw=0 planes (p=0,3,6): column 131  (lw=128 OOB tap, in pad-skip region)
    //   kw=2 planes (p=2,5,8): column 4    (lw=-1  OOB tap, before DMA start)
    if (t < 48u) {
        unsigned g = t >> 3, r = t & 7u;
        unsigned p   = (g < 3u) ? (g * 3u) : ((g - 3u) * 3u + 2u);
        unsigned col = (g < 3u) ? 131u : 4u;
        lds[p * PLANE_F + r * PITCH + col] = 0.0f;
    }
    // top edge: kh==2 planes (p=6,7,8) want source row -1 -> zero LDS row 0
    if (h0 == 0) {
        for (unsigned i = t; i < 3u * PITCH; i += 256u)
            lds[(6u + i / PITCH) * PLANE_F + (i % PITCH)] = 0.0f;
    }
    // bottom edge: kh==0 planes (p=0,1,2) want source row 128 -> zero last LDS row
    if (h0 == HH - ROWS) {
        for (unsigned i = t; i < 3u * PITCH; i += 256u)
            lds[(i / PITCH) * PLANE_F + (ROWS - 1) * PITCH + (i % PITCH)] = 0.0f;
    }

    // ---- wave 0 issues 9 TDM tensor_load_to_lds ops (EXEC-independent, once/wave) ----
    if (t < 32u) {
        const uint64_t xbase = (uint64_t)(uintptr_t)x +
            4ull * ((uint64_t)(b * (CC * NPLANE) + c * NPLANE) * (LHD * LWD));
        const unsigned ldsbase = (unsigned)(uintptr_t)(&lds[0]);
        #pragma unroll
        for (int kh = 0; kh < 3; ++kh) {
            int s  = h0 + 1 - kh;                 // first source row needed (may be -1)
            int lo = (s < 0) ? -s : 0;            // rows clipped at top (0 or 1)
            int sc = s + lo;                      // clipped start row
            int hi = (sc + (ROWS - lo) > LHD) ? (sc + (ROWS - lo) - LHD) : 0;
            int nrows = ROWS - lo - hi;           // rows actually transferred (7 or 8)
            #pragma unroll
            for (int kw = 0; kw < 3; ++kw) {
                int p = kh * 3 + kw;
                uint64_t ga = xbase + 4ull * ((uint64_t)p * (LHD * LWD) + (uint64_t)sc * LWD);
                unsigned la = ldsbase +
                    4u * (unsigned)(p * PLANE_F + lo * PITCH + (kw + 3));

                // ---- D# group 0 (128b): count / lds_addr / 57-bit global_addr / type=2
                v4u g0;
                g0.x = 1u;                                  // count = 1 (valid descriptor)
                g0.y = la;                                  // lds_addr (bytes)
                g0.z = (unsigned)ga;                        // global_addr[31:0]
                g0.w = (unsigned)(ga >> 32) | 0x80000000u;  // global_addr[56:32] | type=2<<30

                // ---- D# group 1 (256b): sizes / dims / strides / LDS pad
                v8u g1;
                g1.s0 = (2u << 16)      // data_size = 4 bytes
                      | (1u << 20)      // pad_enable
                      | (6u << 22)      // pad_interval: pad after 128 dwords (one row)
                      | (7u << 25);     // pad_amount: 8 dwords -> LDS pitch 136 floats
                g1.s1 = ((unsigned)LWD   & 0xFFFFu) << 16;  // tensor_dim0[15:0] = 128
                g1.s2 = ((unsigned)nrows & 0xFFFFu) << 16;  // tensor_dim0 hi=0 | tensor_dim1 lo
                g1.s3 = ((unsigned)LWD   & 0xFFFFu) << 16;  // tensor_dim1 hi=0 | tile_dim0 = 128
                g1.s4 = (unsigned)nrows  & 0xFFFFu;         // tile_dim1 = nrows | tile_dim2 = 0
                g1.s5 = (unsigned)LWD;                      // tensor_dim0_stride = 128 elements
                g1.s6 = 0u;                                 // stride hi / tensor_dim1_stride lo
                g1.s7 = 0u;                                 // tensor_dim1_stride hi

                // 2-D tensor: groups 2/3 are NULL -> two-operand form
                asm volatile("tensor_load_to_lds %0, %1" :: "s"(g0), "s"(g1) : "memory");
            }
        }
        __builtin_amdgcn_s_wait_tensorcnt(0);
    }
    __syncthreads();

    // ---- compute: thread t owns row hh = t>>5, cols w0..w0+3 (w0 = (t&31)*4) ----
    // 9 aligned ds_load_b128 (same column w0+4 in every plane) + 1 global_store_b128.
    {
        const unsigned hh = t >> 5;
        const unsigned w0 = (t & 31u) << 2;
        v4f acc = {0.0f, 0.0f, 0.0f, 0.0f};
        #pragma unroll
        for (int p = 0; p < NPLANE; ++p) {
            const v4f* src = (const v4f*)&lds[p * PLANE_F + hh * PITCH + w0 + 4u];
            acc += *src;
        }
        v4f* op = (v4f*)(out + ((size_t)(b * CC + c) * HH + (size_t)(h0 + (int)hh)) * WW + w0);
        __builtin_nontemporal_store(acc, op);
    }
}

extern "C" void kernel_launch(void* const* d_in, const int* in_sizes, int n_in,
                              void* d_out, int out_size, void* d_ws, size_t ws_size,
                              hipStream_t stream) {
    (void)in_sizes; (void)n_in; (void)d_ws; (void)ws_size; (void)out_size;
    const float* x = (const float*)d_in[0];
    float* out = (float*)d_out;
    // 16 b * 64 c * 16 row-tiles = 16384 blocks, 256 threads (8 waves)
    fold_tdm_kernel<<<dim3(16384), dim3(256), 0, stream>>>(x, out);
}